// ContentEncoder_4260607558279
// MI455X (gfx1250) — compile-verified
//
#include <hip/hip_runtime.h>
#include <cstdint>

// ---------------------------------------------------------------------------
// CDNA5 (gfx1250) ContentEncoder:
//   4x modulated-deformable-conv (+InstanceNorm+ReLU) stages, then 2 residual
//   blocks at 8x256x32x32.  All GEMMs run on the WMMA pipe via
//   V_WMMA_F32_16X16X4_F32 (fp32 in / fp32 accumulate).
//
// GEMM reduction is reordered as s = p*CINP + ci so all deformable-sampling
// setup (offset/mask loads, sigmoid, floor, bilinear weights, validity) is
// hoisted to once per kernel-tap p; the inner channel loop is pure gathers +
// WMMA.  Each wave computes a 32(Cout) x 64(positions) tile: MT=2 oc
// fragments x NT=4 position fragments = 8 v_wmma per 4-deep K step.
// ---------------------------------------------------------------------------

typedef __attribute__((ext_vector_type(2))) float v2f;
typedef __attribute__((ext_vector_type(8))) float v8f;

#define IN_EPS 1e-5f

__device__ __forceinline__ v8f wmma_f32_16x16x4(v2f a, v2f b, v8f c) {
    // 8 args: (neg_a, A, neg_b, B, c_mod, C, reuse_a, reuse_b)
    return __builtin_amdgcn_wmma_f32_16x16x4_f32(false, a, false, b,
                                                 (short)0, c, false, false);
}

// Fragment layouts (f32 16x16x4, wave32):
//   A (16x4): lane<16 -> K=0,1 ; lane>=16 -> K=2,3 ; M = lane&15
//   B (4x16): vgpr j, lane half h -> K = 2*h + j ; N = lane&15
//   C/D     : vgpr v -> M = v + 8*half ; N = lane&15
//
// Geometry (all reference layers): Wo = Ho = 1<<LOGWO ; H = W = Wo*STRIDE ;
// HW divisible by 64.
template <int CIN, int K, int STRIDE, int PAD, int LOGWO, int COUT,
          bool DEFORM>
__global__ void conv_wmma_kernel(const float* __restrict__ x,
                                 const float* __restrict__ w,
                                 const float* __restrict__ bias,
                                 const float* __restrict__ om,
                                 float* __restrict__ out, int B) {
    constexpr int Wo = 1 << LOGWO;
    constexpr int HW = Wo * Wo;
    constexpr int H  = Wo * STRIDE;
    constexpr int W  = H;
    constexpr int KK = K * K;
    constexpr int R  = CIN * KK;           // true reduction depth
    constexpr int CINP = (CIN + 3) & ~3;   // padded channel count (slices of 4)
    constexpr int NT = 4;                  // position sub-tiles (64 positions)
    constexpr int MT = 2;                  // oc sub-tiles (32 out channels)
    constexpr int NCOR = DEFORM ? 4 : 1;   // gather corners per sample
    constexpr int nOcSup = (COUT + 16 * MT - 1) / (16 * MT);
    constexpr int nPos   = HW / (16 * NT);
    constexpr int tilesPerB = nOcSup * nPos;
    constexpr bool OC_GUARD = (COUT % (16 * MT)) != 0;
    constexpr bool CI_GUARD = (CINP != CIN);

    int wave = blockIdx.x * ((int)blockDim.x >> 5) + ((int)threadIdx.x >> 5);
    if (wave >= B * tilesPerB) return;               // wave-uniform exit

    int b      = wave / tilesPerB;
    int rem    = wave - b * tilesPerB;
    int octile = rem / nPos;
    int ptile  = rem - octile * nPos;
    int oc0    = octile * (16 * MT);
    int n0     = ptile * (16 * NT);

    const int lane = (int)threadIdx.x & 31;
    const int half = lane >> 4;
    const int m    = lane & 15;

    // A-side: one weight row pointer per oc sub-tile
    bool aok[MT];
    const float* wrow[MT];
#pragma unroll
    for (int u = 0; u < MT; ++u) {
        int ocA = oc0 + u * 16 + m;
        aok[u]  = OC_GUARD ? (ocA < COUT) : true;
        wrow[u] = w + (size_t)(aok[u] ? ocA : 0) * R;
    }

    // B-side: per-sub-tile column indices (hw == n; Wo is a power of two)
    int nc[NT], oyc[NT], oxc[NT];
#pragma unroll
    for (int t = 0; t < NT; ++t) {
        nc[t]  = n0 + t * 16 + m;
        oyc[t] = nc[t] >> LOGWO;
        oxc[t] = nc[t] & (Wo - 1);
    }
    const float* xb   = x + (size_t)b * CIN * H * W;
    const float* om_b = DEFORM ? (om + (size_t)b * 3 * KK * HW) : nullptr;

    v8f acc[MT][NT];
#pragma unroll
    for (int u = 0; u < MT; ++u)
#pragma unroll
        for (int t = 0; t < NT; ++t)
            acc[u][t] = (v8f){0.f, 0.f, 0.f, 0.f, 0.f, 0.f, 0.f, 0.f};

    for (int p = 0; p < KK; ++p) {
        const int ky = p / K;          // compile-time divisor
        const int kx = p - ky * K;

        // ---- per-tap precompute: (weight, offset) pairs, validity and the
        //      modulation mask folded into the weights ----
        float wgt[NT][NCOR];
        int   ofs[NT][NCOR];
#pragma unroll
        for (int t = 0; t < NT; ++t) {
            if (DEFORM) {
                // DCNv2 channel map: dy = om[2p], dx = om[2p+1],
                // mask = sigmoid(om[2*KK + p]).
                float dy = om_b[(2 * p) * HW + nc[t]];
                float dx = om_b[(2 * p + 1) * HW + nc[t]];
                float mk = om_b[(2 * KK + p) * HW + nc[t]];
                mk = 1.0f / (1.0f + __expf(-mk));

                float py = (float)(oyc[t] * STRIDE - PAD + ky) + dy;
                float px = (float)(oxc[t] * STRIDE - PAD + kx) + dx;
                float fy0 = floorf(py), fx0 = floorf(px);
                float wy = py - fy0, wx = px - fx0;
                int iy0 = (int)fy0, ix0 = (int)fx0;
                int iy1 = iy0 + 1, ix1 = ix0 + 1;
                bool vy0 = (iy0 >= 0) & (iy0 < H);
                bool vy1 = (iy1 >= 0) & (iy1 < H);
                bool vx0 = (ix0 >= 0) & (ix0 < W);
                bool vx1 = (ix1 >= 0) & (ix1 < W);
                int cy0 = iy0 < 0 ? 0 : (iy0 > H - 1 ? H - 1 : iy0);
                int cy1 = iy1 < 0 ? 0 : (iy1 > H - 1 ? H - 1 : iy1);
                int cx0 = ix0 < 0 ? 0 : (ix0 > W - 1 ? W - 1 : ix0);
                int cx1 = ix1 < 0 ? 0 : (ix1 > W - 1 ? W - 1 : ix1);
                wgt[t][0] = (vy0 & vx0) ? (1.f - wy) * (1.f - wx) * mk : 0.f;
                wgt[t][1] = (vy0 & vx1) ? (1.f - wy) * wx * mk : 0.f;
                wgt[t][2] = (vy1 & vx0) ? wy * (1.f - wx) * mk : 0.f;
                wgt[t][3] = (vy1 & vx1) ? wy * wx * mk : 0.f;
                ofs[t][0] = cy0 * W + cx0;
                ofs[t][1] = cy0 * W + cx1;
                ofs[t][2] = cy1 * W + cx0;
                ofs[t][3] = cy1 * W + cx1;
            } else {
                int iy = oyc[t] * STRIDE - PAD + ky;
                int ix = oxc[t] * STRIDE - PAD + kx;
                bool ok = (iy >= 0) & (iy < H) & (ix >= 0) & (ix < W);
                int cy = iy < 0 ? 0 : (iy > H - 1 ? H - 1 : iy);
                int cx = ix < 0 ? 0 : (ix > W - 1 ? W - 1 : ix);
                wgt[t][0] = ok ? 1.0f : 0.0f;
                ofs[t][0] = cy * W + cx;
            }
        }
        if (DEFORM && p + 1 < KK)      // gfx1250 global_prefetch_b8
            __builtin_prefetch(&om_b[(2 * (p + 1)) * HW + nc[0]], 0, 1);

        // ---- inner channel loop: pure gathers + WMMA ----
        for (int c0 = 0; c0 < CINP; c0 += 4) {
            v2f av[MT];
            v2f bv[NT];
#pragma unroll
            for (int j = 0; j < 2; ++j) {
                int ci = c0 + 2 * half + j;
                bool cok = CI_GUARD ? (ci < CIN) : true;
                int cic  = CI_GUARD ? (cok ? ci : CIN - 1) : ci;
#pragma unroll
                for (int u = 0; u < MT; ++u) {
                    bool ok = aok[u] && cok;
                    av[u][j] = ok ? wrow[u][cic * KK + p] : 0.0f;
                }
                const float* xc = xb + (size_t)cic * H * W;
#pragma unroll
                for (int t = 0; t < NT; ++t) {
                    float s = 0.0f;
#pragma unroll
                    for (int cor = 0; cor < NCOR; ++cor)
                        s += wgt[t][cor] * xc[ofs[t][cor]];
                    bv[t][j] = s;    // zero A column nullifies padded slots
                }
            }
#pragma unroll
            for (int u = 0; u < MT; ++u)
#pragma unroll
                for (int t = 0; t < NT; ++t)
                    acc[u][t] = wmma_f32_16x16x4(av[u], bv[t], acc[u][t]);
        }
    }

    // ---- store D (M = v + 8*half per fragment) + bias ----
#pragma unroll
    for (int u = 0; u < MT; ++u) {
#pragma unroll
        for (int v = 0; v < 8; ++v) {
            int oc = oc0 + u * 16 + v + 8 * half;
            if (!OC_GUARD || oc < COUT) {
                float bval  = bias[oc];
                float* orow = out + ((size_t)b * COUT + oc) * HW;
#pragma unroll
                for (int t = 0; t < NT; ++t)
                    orow[nc[t]] = acc[u][t][v] + bval;
            }
        }
    }
}

// ---------------------------------------------------------------------------
// Instance norm: per-(b,c) mean / rsqrt(var+eps); normalize (+ReLU) in place,
// or fused residual update h += 0.1 * norm(r2).
// ---------------------------------------------------------------------------
__global__ void in_stats_kernel(const float* __restrict__ x,
                                float* __restrict__ mean,
                                float* __restrict__ inv, int HW) {
    int bc = blockIdx.x;
    const float* p = x + (long)bc * HW;
    float s = 0.f, s2 = 0.f;
    for (int i = threadIdx.x; i < HW; i += blockDim.x) {
        float v = p[i];
        s += v;
        s2 += v * v;
    }
    __shared__ float sh[256];
    __shared__ float sh2[256];
    sh[threadIdx.x] = s;
    sh2[threadIdx.x] = s2;
    __syncthreads();
    for (int off = blockDim.x >> 1; off > 0; off >>= 1) {
        if ((int)threadIdx.x < off) {
            sh[threadIdx.x] += sh[threadIdx.x + off];
            sh2[threadIdx.x] += sh2[threadIdx.x + off];
        }
        __syncthreads();
    }
    if (threadIdx.x == 0) {
        float mu = sh[0] / (float)HW;
        float va = fmaxf(sh2[0] / (float)HW - mu * mu, 0.0f);
        mean[bc] = mu;
        inv[bc] = rsqrtf(va + IN_EPS);
    }
}

__global__ void in_apply_kernel(float* __restrict__ x,
                                const float* __restrict__ mean,
                                const float* __restrict__ inv,
                                int HW, int total, int do_relu) {
    int i = blockIdx.x * blockDim.x + threadIdx.x;
    if (i >= total) return;
    int bc = i / HW;
    float v = (x[i] - mean[bc]) * inv[bc];
    if (do_relu) v = fmaxf(v, 0.0f);
    x[i] = v;
}

__global__ void res_update_kernel(float* __restrict__ h,
                                  const float* __restrict__ r2,
                                  const float* __restrict__ mean,
                                  const float* __restrict__ inv,
                                  int HW, int total) {
    int i = blockIdx.x * blockDim.x + threadIdx.x;
    if (i >= total) return;
    int bc = i / HW;
    h[i] += 0.1f * ((r2[i] - mean[bc]) * inv[bc]);
}

// ---------------------------------------------------------------------------
// Host orchestration
// ---------------------------------------------------------------------------
template <int CIN, int K, int STRIDE, int PAD, int LOGWO, int COUT,
          bool DEFORM>
static inline void launch_cv(const float* x, const float* w, const float* b,
                             const float* om, float* out, int B,
                             hipStream_t st) {
    constexpr long HW = (long)(1 << LOGWO) * (1 << LOGWO);
    constexpr long nOcSup = (COUT + 31) / 32;
    long waves = (long)B * nOcSup * (HW / 64);
    dim3 blk(128);                       // 4 waves / block (wave32)
    dim3 grd((unsigned)((waves + 3) / 4));
    conv_wmma_kernel<CIN, K, STRIDE, PAD, LOGWO, COUT, DEFORM>
        <<<grd, blk, 0, st>>>(x, w, b, om, out, B);
}

static inline void launch_instnorm(float* x, float* mean, float* inv, int BC,
                                   int HW, int do_relu, hipStream_t st) {
    in_stats_kernel<<<dim3(BC), dim3(256), 0, st>>>(x, mean, inv, HW);
    int total = BC * HW;
    in_apply_kernel<<<dim3((total + 255) / 256), dim3(256), 0, st>>>(
        x, mean, inv, HW, total, do_relu);
}

extern "C" void kernel_launch(void* const* d_in, const int* in_sizes, int n_in,
                              void* d_out, int out_size, void* d_ws,
                              size_t ws_size, hipStream_t stream) {
    const float* x     = (const float*)d_in[0];
    const float* d1w   = (const float*)d_in[1];
    const float* d1b   = (const float*)d_in[2];
    const float* d1ow  = (const float*)d_in[3];
    const float* d1ob  = (const float*)d_in[4];
    const float* d2w   = (const float*)d_in[5];
    const float* d2b   = (const float*)d_in[6];
    const float* d2ow  = (const float*)d_in[7];
    const float* d2ob  = (const float*)d_in[8];
    const float* d3w   = (const float*)d_in[9];
    const float* d3b   = (const float*)d_in[10];
    const float* d3ow  = (const float*)d_in[11];
    const float* d3ob  = (const float*)d_in[12];
    const float* d4w   = (const float*)d_in[13];
    const float* d4b   = (const float*)d_in[14];
    const float* d4ow  = (const float*)d_in[15];
    const float* d4ob  = (const float*)d_in[16];
    const float* res_w = (const float*)d_in[17];
    const float* res_b = (const float*)d_in[18];

    const int B = 8;

    // --- workspace layout ---
    char* ws = (char*)d_ws;
    size_t cur = 0;
    auto alloc = [&](size_t bytes) -> char* {
        char* p = ws + cur;
        cur += (bytes + 255) & ~(size_t)255;
        return p;
    };
    float* h1   = (float*)alloc((size_t)B * 32 * 256 * 256 * 4);   //  67.1 MB
    float* h2   = (float*)alloc((size_t)B * 64 * 128 * 128 * 4);   //  33.6 MB
    float* h3   = (float*)alloc((size_t)B * 128 * 64 * 64 * 4);    //  16.8 MB
    float* omrg = (float*)alloc((size_t)B * 147 * 256 * 256 * 4);  // 308.3 MB
    float* mean = (float*)alloc(2048 * 4);
    float* inv  = (float*)alloc(2048 * 4);
    if (cur > ws_size) return;

    float* h4 = (float*)d_out;                       // 8x256x32x32
    float* rA = omrg;                                // residual temp r
    float* rB = omrg + (size_t)B * 256 * 32 * 32;    // residual temp r2

    // ---- dcn1: 3 -> 32, K=7, s=1, p=3, 256x256 (offC = 147) ----
    launch_cv<3, 7, 1, 3, 8, 147, false>(x, d1ow, d1ob, nullptr, omrg, B,
                                         stream);
    launch_cv<3, 7, 1, 3, 8, 32, true>(x, d1w, d1b, omrg, h1, B, stream);
    launch_instnorm(h1, mean, inv, B * 32, 256 * 256, 1, stream);

    // ---- dcn2: 32 -> 64, K=4, s=2, p=1, -> 128x128 (offC = 48) ----
    launch_cv<32, 4, 2, 1, 7, 48, false>(h1, d2ow, d2ob, nullptr, omrg, B,
                                         stream);
    launch_cv<32, 4, 2, 1, 7, 64, true>(h1, d2w, d2b, omrg, h2, B, stream);
    launch_instnorm(h2, mean, inv, B * 64, 128 * 128, 1, stream);

    // ---- dcn3: 64 -> 128, K=4, s=2, p=1, -> 64x64 ----
    launch_cv<64, 4, 2, 1, 6, 48, false>(h2, d3ow, d3ob, nullptr, omrg, B,
                                         stream);
    launch_cv<64, 4, 2, 1, 6, 128, true>(h2, d3w, d3b, omrg, h3, B, stream);
    launch_instnorm(h3, mean, inv, B * 128, 64 * 64, 1, stream);

    // ---- dcn4: 128 -> 256, K=4, s=2, p=1, -> 32x32 ----
    launch_cv<128, 4, 2, 1, 5, 48, false>(h3, d4ow, d4ob, nullptr, omrg, B,
                                          stream);
    launch_cv<128, 4, 2, 1, 5, 256, true>(h3, d4w, d4b, omrg, h4, B, stream);
    launch_instnorm(h4, mean, inv, B * 256, 32 * 32, 1, stream);

    // ---- residual blocks (C=256, 32x32, K=3, s=1, p=1) ----
    const int C = 256, HWr = 32 * 32;
    const size_t wblk = (size_t)C * C * 9;
    for (int i = 0; i < 2; ++i) {
        const float* w0 = res_w + (size_t)i * 2 * wblk;
        const float* w1 = w0 + wblk;
        const float* b0 = res_b + (size_t)i * 2 * C;
        const float* b1 = b0 + C;

        launch_cv<256, 3, 1, 1, 5, 256, false>(h4, w0, b0, nullptr, rA, B,
                                               stream);
        launch_instnorm(rA, mean, inv, B * C, HWr, 1, stream);

        launch_cv<256, 3, 1, 1, 5, 256, false>(rA, w1, b1, nullptr, rB, B,
                                               stream);
        in_stats_kernel<<<dim3(B * C), dim3(256), 0, stream>>>(rB, mean, inv,
                                                               HWr);
        int total = B * C * HWr;
        res_update_kernel<<<dim3((total + 255) / 256), dim3(256), 0, stream>>>(
            h4, rB, mean, inv, HWr, total);
    }
}